// MultiHeadAttention_53635551592813
// MI455X (gfx1250) — compile-verified
//
#include <hip/hip_runtime.h>
#include <hip/hip_bf16.h>
#include <math.h>

// Problem constants (start_pos fixed at 4095 in the reference)
constexpr int BS      = 16;
constexpr int DIM     = 4096;
constexpr int H       = 32;
constexpr int DQK     = 128;
constexpr int DV      = 128;
constexpr int KVLEN   = 4096;   // start_pos + 1
constexpr int NEWPOS  = 4095;   // start_pos
constexpr float SCALE = 0.08838834764831845f; // 1/sqrt(128)

typedef __attribute__((ext_vector_type(2))) float v2f;
typedef __attribute__((ext_vector_type(4))) float v4f;
typedef __attribute__((ext_vector_type(8))) float v8f;

// ---------------------------------------------------------------------------
// C[16,N] = A[16,K] @ B[K,N], fp32, via V_WMMA_F32_16X16X4_F32.
// One wave owns a 16x16 tile. K,N compile-time -> pure pointer-increment
// streams with immediate offsets; B is stream-once -> NT loads + prefetch.
// ---------------------------------------------------------------------------
template <int K, int N>
__global__ void __launch_bounds__(256)
gemm16_wmma_f32(const float* __restrict__ A, const float* __restrict__ B,
                float* __restrict__ C) {
  const int wave = threadIdx.x >> 5;
  const int lane = threadIdx.x & 31;
  const int half = lane >> 4;        // 0: lanes 0-15, 1: lanes 16-31
  const int l16  = lane & 15;
  const int colBase = (blockIdx.x * 8 + wave) * 16;

  // A layout (16x4 f32): lanes 0-15 hold K={k0,k0+1}, lanes 16-31 K={k0+2,k0+3}
  const float* __restrict__ ap = A + (size_t)l16 * K + 2 * half;
  // B layout (4x16 f32): VGPR0 rows {k0, k0+2}, VGPR1 rows {k0+1, k0+3}
  const float* __restrict__ bp = B + (size_t)(2 * half) * N + colBase + l16;

  v8f acc = {};
  #pragma unroll 4
  for (int k0 = 0; k0 < K; k0 += 4) {
    v2f a, b;
    a.x = ap[0];
    a.y = ap[1];
    b.x = __builtin_nontemporal_load(bp);
    b.y = __builtin_nontemporal_load(bp + N);
    __builtin_prefetch(bp + 16 * (size_t)N, 0, 1);  // ~4 iters ahead, immediate offset
    // 8 args: (neg_a, A, neg_b, B, c_mod, C, reuse_a, reuse_b)
    acc = __builtin_amdgcn_wmma_f32_16x16x4_f32(
        false, a, false, b, (short)0, acc, false, false);
    ap += 4;
    bp += (size_t)4 * N;
  }

  #pragma unroll
  for (int i = 0; i < 8; ++i) {
    const int row = i + 8 * half;    // C/D layout: VGPR i -> rows {i, i+8}
    __builtin_nontemporal_store(acc[i], C + (size_t)row * N + colBase + l16);
  }
}

// ---------------------------------------------------------------------------
// Fused decode attention, one (b,h) per block (256 thr = 8 waves).
// Each position's 128-dim fp32 vector = 512 contiguous bytes = one float4 per
// lane across wave32 -> perfectly coalesced NT streaming of the 2 GB KV cache.
// Position 4095 reads the freshly projected xk/xv (cache is NOT mutated).
// ---------------------------------------------------------------------------
__global__ void __launch_bounds__(256)
attn_decode(const float* __restrict__ xq, const float* __restrict__ xk,
            const float* __restrict__ xv, const float* __restrict__ cache_k,
            const float* __restrict__ cache_v, const float* __restrict__ mask,
            float* __restrict__ weight_out, float* __restrict__ ctx_out) {
  const int h = blockIdx.x;
  const int b = blockIdx.y;
  const int tid  = threadIdx.x;
  const int wave = tid >> 5;
  const int lane = tid & 31;

  __shared__ float s_scores[KVLEN];
  __shared__ float s_red[256];
  __shared__ float s_ctx[DV];

  if (tid < DV) s_ctx[tid] = 0.0f;

  // q for this (b,h): lane owns dims [4*lane, 4*lane+3]
  const float* qp = xq + (size_t)b * (H * DQK) + h * DQK + lane * 4;
  const float q0 = qp[0], q1 = qp[1], q2 = qp[2], q3 = qp[3];

  const float* knew = xk + (size_t)b * (H * DQK) + h * DQK;
  const float* vnew = xv + (size_t)b * (H * DV) + h * DV;
  float* wout = weight_out + ((size_t)b * H + h) * KVLEN;

  const size_t pstep = (size_t)8 * H * DQK;  // 8 positions of cache stride

  // ---- scores for pos 0..4094 from cache (pointer-increment stream) ----
  {
    const float* kp =
        cache_k + (((size_t)b * KVLEN + wave) * H + h) * DQK + lane * 4;
    #pragma unroll 2
    for (int pos = wave; pos < NEWPOS; pos += 8, kp += pstep) {
      __builtin_prefetch(kp + 4 * pstep, 0, 1);
      const v4f kv = __builtin_nontemporal_load((const v4f*)kp);
      float part = q0 * kv.x + q1 * kv.y + q2 * kv.z + q3 * kv.w;
      #pragma unroll
      for (int off = 16; off >= 1; off >>= 1)
        part += __shfl_xor(part, off, 32);
      if (lane == 0) s_scores[pos] = part * SCALE + mask[pos];
    }
  }
  // ---- pos 4095 from fresh xk (4095 % 8 == 7 -> wave 7) ----
  if (wave == 7) {
    const v4f kv = *(const v4f*)(knew + lane * 4);
    float part = q0 * kv.x + q1 * kv.y + q2 * kv.z + q3 * kv.w;
    #pragma unroll
    for (int off = 16; off >= 1; off >>= 1)
      part += __shfl_xor(part, off, 32);
    if (lane == 0) s_scores[NEWPOS] = part * SCALE + mask[NEWPOS];
  }
  __syncthreads();

  // ---- block max; also emit pre-softmax "weight" (coalesced NT stores) ----
  float lmax = -INFINITY;
  for (int p = tid; p < KVLEN; p += 256) {
    const float s = s_scores[p];
    __builtin_nontemporal_store(s, wout + p);
    lmax = fmaxf(lmax, s);
  }
  s_red[tid] = lmax;
  __syncthreads();
  for (int s = 128; s > 0; s >>= 1) {
    if (tid < s) s_red[tid] = fmaxf(s_red[tid], s_red[tid + s]);
    __syncthreads();
  }
  const float gmax = s_red[0];
  __syncthreads();

  // ---- exp + block sum ----
  float lsum = 0.0f;
  for (int p = tid; p < KVLEN; p += 256) {
    const float e = __expf(s_scores[p] - gmax);
    s_scores[p] = e;
    lsum += e;
  }
  s_red[tid] = lsum;
  __syncthreads();
  for (int s = 128; s > 0; s >>= 1) {
    if (tid < s) s_red[tid] += s_red[tid + s];
    __syncthreads();
  }
  const float inv = 1.0f / s_red[0];
  __syncthreads();

  // ---- ctx = sum_pos attn[pos] * v[pos] ----
  float a0 = 0.f, a1 = 0.f, a2 = 0.f, a3 = 0.f;
  {
    const float* vp =
        cache_v + (((size_t)b * KVLEN + wave) * H + h) * DV + lane * 4;
    #pragma unroll 2
    for (int pos = wave; pos < NEWPOS; pos += 8, vp += pstep) {
      __builtin_prefetch(vp + 4 * pstep, 0, 1);
      const v4f vv = __builtin_nontemporal_load((const v4f*)vp);
      const float w = s_scores[pos];
      a0 += w * vv.x; a1 += w * vv.y; a2 += w * vv.z; a3 += w * vv.w;
    }
  }
  if (wave == 7) {
    const v4f vv = *(const v4f*)(vnew + lane * 4);
    const float w = s_scores[NEWPOS];
    a0 += w * vv.x; a1 += w * vv.y; a2 += w * vv.z; a3 += w * vv.w;
  }
  atomicAdd(&s_ctx[lane * 4 + 0], a0);   // ds_add_f32
  atomicAdd(&s_ctx[lane * 4 + 1], a1);
  atomicAdd(&s_ctx[lane * 4 + 2], a2);
  atomicAdd(&s_ctx[lane * 4 + 3], a3);
  __syncthreads();

  if (tid < DV)
    ctx_out[(size_t)b * (H * DV) + h * DV + tid] = s_ctx[tid] * inv;
}

// ---------------------------------------------------------------------------
// Launch: 3 WMMA GEMMs (q,k,v) -> attention -> WMMA GEMM (wo)
// ---------------------------------------------------------------------------
extern "C" void kernel_launch(void* const* d_in, const int* in_sizes, int n_in,
                              void* d_out, int out_size, void* d_ws, size_t ws_size,
                              hipStream_t stream) {
  const float* x       = (const float*)d_in[0];  // [16,1,4096]
  const float* mask    = (const float*)d_in[1];  // [1,1,1,4096]
  const float* wq      = (const float*)d_in[2];  // [4096,4096]
  const float* wk      = (const float*)d_in[3];
  const float* wv      = (const float*)d_in[4];
  const float* wo      = (const float*)d_in[5];
  const float* cache_k = (const float*)d_in[6];  // [16,4096,32,128]
  const float* cache_v = (const float*)d_in[7];

  float* ws  = (float*)d_ws;
  float* xq  = ws;                    // [16, 4096]
  float* xk  = ws + 1 * BS * DIM;     // [16, 4096]
  float* xv  = ws + 2 * BS * DIM;     // [16, 4096]
  float* ctx = ws + 3 * BS * DIM;     // [16, 4096]

  float* weight_out = (float*)d_out;                          // 16*32*4096
  float* output     = (float*)d_out + (size_t)BS * H * KVLEN; // 16*4096

  // All four GEMMs are [16,4096] x [4096,4096]: one instantiation.
  dim3 gblk(256);
  dim3 ggrd(DIM / 16 / 8);   // 256 tiles / 8 waves-per-block
  gemm16_wmma_f32<DIM, DIM><<<ggrd, gblk, 0, stream>>>(x, wq, xq);
  gemm16_wmma_f32<DIM, DIM><<<ggrd, gblk, 0, stream>>>(x, wk, xk);
  gemm16_wmma_f32<DIM, DIM><<<ggrd, gblk, 0, stream>>>(x, wv, xv);

  // Attention: one block per (b,h)
  dim3 agrd(H, BS);
  attn_decode<<<agrd, dim3(256), 0, stream>>>(xq, xk, xv, cache_k, cache_v,
                                              mask, weight_out, ctx);

  // Output projection
  gemm16_wmma_f32<DIM, DIM><<<ggrd, gblk, 0, stream>>>(ctx, wo, output);
}